// PacketTimeLSTM_3_31190052504104
// MI455X (gfx1250) — compile-verified
//
#include <hip/hip_runtime.h>
#include <hip/hip_bf16.h>
#include <math.h>

#define F_FEAT 4096
#define HID    128
#define GATES  384          // 3*HID
#define INW    129          // HID+1
#define NBLK   1024         // blocks in LSTM kernel
#define FPB    4            // features per block (NBLK*FPB == F_FEAT)

typedef float v2f __attribute__((ext_vector_type(2)));
typedef float v4f __attribute__((ext_vector_type(4)));
typedef float v8f __attribute__((ext_vector_type(8)));

// d_ws layout (floats):
//   [0] cnt = max(#mask,1)   [1] anyMask flag
//   [8 .. 264)   feat_in  (C_curr | H_agg)
//   [512 .. 512 + NBLK*256)  per-block partials (128 C-sum | 128 H-sum)
#define WS_FEAT 8
#define WS_PART 512

__device__ __forceinline__ float sigf(float x) { return 1.0f / (1.0f + expf(-x)); }

// ---------------------------------------------------------------------------
// Kernel 0: count mask (cnt = max(sum,1)), anyMask flag
// ---------------------------------------------------------------------------
__global__ __launch_bounds__(256) void k_mask_count(const int* __restrict__ mask,
                                                    float* __restrict__ ws) {
    __shared__ float s[256];
    const int t = threadIdx.x;
    float c = 0.f;
    for (int i = t; i < F_FEAT; i += 256) c += (mask[i] != 0) ? 1.f : 0.f;
    s[t] = c;
    __syncthreads();
    for (int off = 128; off > 0; off >>= 1) {
        if (t < off) s[t] += s[t + off];
        __syncthreads();
    }
    if (t == 0) {
        ws[0] = fmaxf(s[0], 1.f);
        ws[1] = (s[0] > 0.f) ? 1.f : 0.f;
    }
}

// ---------------------------------------------------------------------------
// Kernel 1: the big one. One 128-thread block handles FPB features.
// Thread t computes mat-vec rows {t, 128+t, 256+t} of W_f (384x129) -> the
// three gate pre-activations of hidden unit t. Streams 812 MB of weights with
// aligned nontemporal b128 loads (head/tail peel for the 129-stride rows).
// ---------------------------------------------------------------------------
__global__ __launch_bounds__(128) void k_lstm_cell(
    const int* __restrict__ timp, const float* __restrict__ X,
    const int* __restrict__ mask, const float* __restrict__ Ht,
    const int* __restrict__ last_occured, const float* __restrict__ c_t,
    const float* __restrict__ W, const float* __restrict__ bias,
    const float* __restrict__ xTw, const float* __restrict__ xTb,
    const float* __restrict__ dTw,
    const float* __restrict__ cinw, const float* __restrict__ coutw,
    const float* __restrict__ ws, float* __restrict__ Hout,
    float* __restrict__ part)
{
    __shared__ float xs[INW];
    const int t = threadIdx.x;
    const int tim = timp[0];
    const float anyMask = ws[1];
    const float ct = c_t[t];
    float csum = 0.f, hsum = 0.f;

    for (int j = 0; j < FPB; ++j) {
        const int f = blockIdx.x * FPB + j;
        const float Xf = X[f];
        if (t == 0) xs[0] = Xf;
        xs[1 + t] = Ht[(size_t)f * HID + t];
        __syncthreads();

        const size_t fg = (size_t)f * GATES;
        const size_t f2 = (size_t)f * 2 * HID;
        const float xm1 = xTw[f2 + t] * Xf + xTb[f2 + t];
        const float xm2 = xTw[f2 + HID + t] * Xf + xTb[f2 + HID + t];
        const float delta = (float)(tim - last_occured[f]);
        const float wd1 = dTw[fg + t];
        float wd2 = dTw[fg + HID + t];
        if (t == 0) wd2 = fminf(wd2, -0.01f);   // boundary clamp on row index H
        const float wdo = dTw[fg + 2 * HID + t];

        const float* w0 = W + (fg + t) * (size_t)INW;
        const float* w1 = w0 + (size_t)HID * INW;
        const float* w2 = w0 + (size_t)(2 * HID) * INW;
        float a0 = bias[fg + t];
        float a1 = bias[fg + HID + t];
        float a2 = bias[fg + 2 * HID + t];

        // row offset in floats == t (mod 4); peel to 16B alignment
        int i = 0;
        const int head = (4 - (t & 3)) & 3;
        for (; i < head; ++i) {
            const float xv = xs[i];
            a0 = fmaf(w0[i], xv, a0);
            a1 = fmaf(w1[i], xv, a1);
            a2 = fmaf(w2[i], xv, a2);
        }
        for (; i + 4 <= INW; i += 4) {
            const v4f v0 = __builtin_nontemporal_load((const v4f*)(w0 + i));
            const v4f v1 = __builtin_nontemporal_load((const v4f*)(w1 + i));
            const v4f v2 = __builtin_nontemporal_load((const v4f*)(w2 + i));
            const float x0 = xs[i], x1 = xs[i + 1], x2 = xs[i + 2], x3 = xs[i + 3];
            a0 = fmaf(v0.x, x0, fmaf(v0.y, x1, fmaf(v0.z, x2, fmaf(v0.w, x3, a0))));
            a1 = fmaf(v1.x, x0, fmaf(v1.y, x1, fmaf(v1.z, x2, fmaf(v1.w, x3, a1))));
            a2 = fmaf(v2.x, x0, fmaf(v2.y, x1, fmaf(v2.z, x2, fmaf(v2.w, x3, a2))));
        }
        for (; i < INW; ++i) {
            const float xv = xs[i];
            a0 = fmaf(w0[i], xv, a0);
            a1 = fmaf(w1[i], xv, a1);
            a2 = fmaf(w2[i], xv, a2);
        }

        // gate math (all in registers, hidden unit t)
        const float T1 = sigf(xm1 + sigf(wd1 * delta));
        const float T2 = sigf(xm2 + sigf(wd2 * delta));
        const float gi = sigf(a0 + cinw[(size_t)f * HID + t] * ct);
        const float gc = tanhf(a2);
        const float giT1 = gi * T1;
        const float c_tilde = (1.f - giT1) * ct + giT1 * gc;
        const float c_new   = (1.f - gi) * ct + gi * T2 * gc;
        const float go = sigf(a1 + coutw[(size_t)f * HID + t] * c_tilde + wdo * delta);
        const float mf = (mask[f] != 0) ? 1.f : 0.f;
        const float hcore = go * tanhf(c_tilde);

        Hout[(size_t)f * HID + t] =
            (anyMask > 0.5f) ? (mf * hcore) : Ht[(size_t)f * HID + t];
        csum += mf * c_new;
        hsum += mf * hcore;   // (mf * H_curr) == mf * hcore in both branches

        __syncthreads();      // xs reused next iteration
    }

    part[(size_t)blockIdx.x * 256 + t] = csum;
    part[(size_t)blockIdx.x * 256 + HID + t] = hsum;
}

// ---------------------------------------------------------------------------
// Kernel 2: deterministic tree-free reduction of block partials -> feat_in,
// C_curr (coalesced: step b reads contiguous 256 floats)
// ---------------------------------------------------------------------------
__global__ __launch_bounds__(256) void k_reduce(const float* __restrict__ part,
                                                const float* __restrict__ ws,
                                                float* __restrict__ feat,
                                                float* __restrict__ Cout) {
    const int t = threadIdx.x;
    float s = 0.f;
    for (int b = 0; b < NBLK; ++b) s += part[(size_t)b * 256 + t];
    s /= ws[0];
    feat[t] = s;                 // [0..128)=C_curr, [128..256)=H_agg
    if (t < HID) Cout[t] = s;
}

// ---------------------------------------------------------------------------
// Kernel 3: MLP head via V_WMMA_F32_16X16X4_F32.
// h1 = relu(W1(256x256) @ feat + b1) as WMMA with B = feat broadcast into all
// 16 N columns (D columns identical). A-tile loaded per ISA layout:
//   lanes 0-15:  row m=mt*16+lane, cols {k0,k0+1}
//   lanes 16-31: row m=mt*16+lane-16, cols {k0+2,k0+3}
// Then pred = softmax(W2 @ h1 + b2).
// ---------------------------------------------------------------------------
__global__ __launch_bounds__(256) void k_mlp(const float* __restrict__ feat,
                                             const float* __restrict__ w1,
                                             const float* __restrict__ b1,
                                             const float* __restrict__ w2,
                                             const float* __restrict__ b2,
                                             float* __restrict__ pred) {
    __shared__ float fs[256];
    __shared__ float h1[256];
    const int t = threadIdx.x;
    fs[t] = feat[t];
    __syncthreads();

    const int wave = t >> 5;
    const int lane = t & 31;
    const int half = lane >> 4;     // 0: K pair {0,1}, 1: K pair {2,3}
    const int m0   = lane & 15;

    for (int mt = wave; mt < 16; mt += 8) {      // 8 waves, 2 M-tiles each
        v8f acc = {0.f, 0.f, 0.f, 0.f, 0.f, 0.f, 0.f, 0.f};
        const int mrow = mt * 16 + m0;
        for (int k0 = 0; k0 < 256; k0 += 4) {
            const v2f a = *(const v2f*)(w1 + (size_t)mrow * 256 + k0 + half * 2);
            v2f b;
            b.x = fs[k0 + half * 2];
            b.y = fs[k0 + half * 2 + 1];
            acc = __builtin_amdgcn_wmma_f32_16x16x4_f32(
                false, a, false, b, (short)0, acc, false, false);
        }
        // D: lane with N==0 holds rows mt*16 + half*8 + v in acc[v]
        if (m0 == 0) {
            #pragma unroll
            for (int v = 0; v < 8; ++v) {
                const int r = mt * 16 + half * 8 + v;
                h1[r] = fmaxf(acc[v] + b1[r], 0.f);
            }
        }
    }
    __syncthreads();

    if (t == 0) {
        float l0 = b2[0], l1 = b2[1];
        for (int i = 0; i < 256; ++i) {
            l0 = fmaf(w2[i], h1[i], l0);
            l1 = fmaf(w2[256 + i], h1[i], l1);
        }
        const float m = fmaxf(l0, l1);
        const float e0 = expf(l0 - m), e1 = expf(l1 - m);
        const float inv = 1.f / (e0 + e1);
        pred[0] = e0 * inv;
        pred[1] = e1 * inv;
    }
}

// ---------------------------------------------------------------------------
extern "C" void kernel_launch(void* const* d_in, const int* in_sizes, int n_in,
                              void* d_out, int out_size, void* d_ws, size_t ws_size,
                              hipStream_t stream) {
    (void)in_sizes; (void)n_in; (void)out_size; (void)ws_size;
    const int*   timp  = (const int*)  d_in[0];
    const float* X     = (const float*)d_in[1];
    /* d_in[2] X_hap unused */
    const int*   mask  = (const int*)  d_in[3];
    const float* Ht    = (const float*)d_in[4];
    /* d_in[5] Ct unused */
    const int*   last  = (const int*)  d_in[6];
    const float* c_t   = (const float*)d_in[7];
    const float* W     = (const float*)d_in[8];
    const float* B     = (const float*)d_in[9];
    const float* xTw   = (const float*)d_in[10];
    const float* xTb   = (const float*)d_in[11];
    const float* dTw   = (const float*)d_in[12];
    const float* cinw  = (const float*)d_in[13];
    const float* coutw = (const float*)d_in[14];
    const float* w1    = (const float*)d_in[15];
    const float* b1    = (const float*)d_in[16];
    const float* w2    = (const float*)d_in[17];
    const float* b2    = (const float*)d_in[18];

    float* out  = (float*)d_out;
    float* pred = out;                              // [0,2)
    float* Hout = out + 2;                          // [2, 2+F*H)
    float* Cout = out + 2 + (size_t)F_FEAT * HID;   // [.., +H)
    float* ws   = (float*)d_ws;
    float* feat = ws + WS_FEAT;
    float* part = ws + WS_PART;

    k_mask_count<<<1, 256, 0, stream>>>(mask, ws);
    k_lstm_cell<<<NBLK, 128, 0, stream>>>(timp, X, mask, Ht, last, c_t, W, B,
                                          xTw, xTb, dTw, cinw, coutw, ws, Hout, part);
    k_reduce<<<1, 256, 0, stream>>>(part, ws, feat, Cout);
    k_mlp<<<1, 256, 0, stream>>>(feat, w1, b1, w2, b2, pred);
}